// LinearAttention_29016799052071
// MI455X (gfx1250) — compile-verified
//
#include <hip/hip_runtime.h>
#include <hip/hip_bf16.h>

// ---------------------------------------------------------------------------
// Linear attention, MI455X (gfx1250, wave32).
// Memory-bound (~0.66 GB traffic vs ~142 GFLOP) -> full-precision FP32 WMMA
// (V_WMMA_F32_16X16X4_F32), async-to-LDS staging for the two big GEMMs.
// ---------------------------------------------------------------------------

typedef __attribute__((ext_vector_type(2))) float v2f;
typedef __attribute__((ext_vector_type(8))) float v8f;

#define DMODEL   1024
#define NHEADS   16
#define HDIM     64
#define BATCH    4
#define SEQ      4096
#define MROWS    (BATCH * SEQ)     // 16384
#define QKV_N    (3 * DMODEL)      // 3072

#define ASTRIDE  36   // floats; 144B rows: 16B-aligned for async B128, bank-conflict-free frags
#define BSTRIDE  72   // floats; half-wave fragment rows hit disjoint bank ranges

__device__ __forceinline__ v8f wmma_f32(v2f a, v2f b, v8f c) {
  // 8 args: (neg_a, A, neg_b, B, c_mod, C, reuse_a, reuse_b)
  return __builtin_amdgcn_wmma_f32_16x16x4_f32(false, a, false, b,
                                               (short)0, c, false, false);
}

// elu(u/scale)+1 with scale = D**0.25 = 64**0.25 = 2^1.5
__device__ __forceinline__ float feature_map(float u) {
  const float inv_scale = 0.35355339059327373f; // 2^-1.5
  float t = u * inv_scale;
  return t > 0.0f ? (t + 1.0f) : __expf(t);
}

__device__ __forceinline__ unsigned lds_off(const void* p) {
  // Generic pointer to __shared__: low 32 bits are the LDS byte offset.
  return (unsigned)(uintptr_t)p;
}

__device__ __forceinline__ void async_ld_b128(unsigned lds_byte, const void* gaddr) {
  asm volatile("global_load_async_to_lds_b128 %0, %1, off"
               :: "v"(lds_byte), "v"((unsigned long long)(uintptr_t)gaddr)
               : "memory");
}
__device__ __forceinline__ void async_ld_b32(unsigned lds_byte, const void* gaddr) {
  asm volatile("global_load_async_to_lds_b32 %0, %1, off"
               :: "v"(lds_byte), "v"((unsigned long long)(uintptr_t)gaddr)
               : "memory");
}
__device__ __forceinline__ void wait_async0() {
  asm volatile("s_wait_asynccnt 0x0" ::: "memory");
}

// ---------------------------------------------------------------------------
// Staged GEMM: C[M,NCOLS] = A[M,1024] @ W[1024,NCOLS] + bias, optional fused
// feature map on the q,k column range (QKV epilogue).
// Block: 256 threads = 8 waves; block tile 128 x 64; each wave 16 rows x 64
// cols (4 WMMA accumulators sharing one A fragment). K staged in 32-chunks:
//   sA[128][32] via 4 async B128/thread (fully coalesced 128B rows)
//   sB[ 32][64] via 8 async B32 /thread (fully coalesced 256B rows)
// grid = (M/128, NCOLS/64)
// ---------------------------------------------------------------------------
template<int NCOLS, bool QKV_EPI>
__global__ __launch_bounds__(256) void gemm_staged(const float* __restrict__ A,
                                                   const float* __restrict__ W,
                                                   const float* __restrict__ bias,
                                                   float* __restrict__ C) {
  __shared__ float sA[128 * ASTRIDE];   // 18 KB
  __shared__ float sB[32 * BSTRIDE];    //  9 KB

  const int tid  = threadIdx.x;
  const int lane = tid & 31;
  const int wave = tid >> 5;
  const int half = lane >> 4;
  const int l    = lane & 15;

  const int m0 = blockIdx.x * 128;
  const int n0 = blockIdx.y * 64;

  // Staging coordinates
  const int arow = tid >> 3, avec = tid & 7;   // A: 8 x vec4 per 32-float row
  const int brow = tid >> 6, bcol = tid & 63;  // B: 64 floats per row

  v8f acc0 = {}, acc1 = {}, acc2 = {}, acc3 = {};

  for (int kc = 0; kc < DMODEL; kc += 32) {
    // ---- async stage A[128][32] ----
#pragma unroll
    for (int p = 0; p < 4; ++p) {
      const int r = arow + p * 32;
      async_ld_b128(lds_off(&sA[r * ASTRIDE + avec * 4]),
                    A + (size_t)(m0 + r) * DMODEL + kc + avec * 4);
    }
    // ---- async stage B[32][64] ----
#pragma unroll
    for (int p = 0; p < 8; ++p) {
      const int r = brow + p * 4;
      async_ld_b32(lds_off(&sB[r * BSTRIDE + bcol]),
                   W + (size_t)(kc + r) * NCOLS + n0 + bcol);
    }
    wait_async0();
    __syncthreads();

    const float* aRow = &sA[(wave * 16 + l) * ASTRIDE];
#pragma unroll
    for (int k = 0; k < 32; k += 4) {
      const int ka = k + 2 * half;
      v2f a = *(const v2f*)(aRow + ka);
      const float* b0p = &sB[ka * BSTRIDE + l];
      const float* b1p = &sB[(ka + 1) * BSTRIDE + l];
      v2f b;
      b.x = b0p[0];  b.y = b1p[0];  acc0 = wmma_f32(a, b, acc0);
      b.x = b0p[16]; b.y = b1p[16]; acc1 = wmma_f32(a, b, acc1);
      b.x = b0p[32]; b.y = b1p[32]; acc2 = wmma_f32(a, b, acc2);
      b.x = b0p[48]; b.y = b1p[48]; acc3 = wmma_f32(a, b, acc3);
    }
    __syncthreads();
  }

  // ---- epilogue ----
  const float bs0 = bias[n0 + l];
  const float bs1 = bias[n0 + 16 + l];
  const float bs2 = bias[n0 + 32 + l];
  const float bs3 = bias[n0 + 48 + l];
  const bool  fm  = QKV_EPI && (n0 < 2 * DMODEL);  // uniform per block (n0 % 64 == 0)

#pragma unroll
  for (int r = 0; r < 8; ++r) {
    const int m = m0 + wave * 16 + r + 8 * half;   // C/D row layout
    float* row = C + (size_t)m * NCOLS + n0;
    float v0 = acc0[r] + bs0;
    float v1 = acc1[r] + bs1;
    float v2 = acc2[r] + bs2;
    float v3 = acc3[r] + bs3;
    if (fm) { v0 = feature_map(v0); v1 = feature_map(v1);
              v2 = feature_map(v2); v3 = feature_map(v3); }
    row[l]      = v0;
    row[16 + l] = v1;
    row[32 + l] = v2;
    row[48 + l] = v3;
  }
}

// ---------------------------------------------------------------------------
// Zero the KV accumulator (split-K uses atomicAdd; ws is not pre-zeroed).
// ---------------------------------------------------------------------------
__global__ void zero_f32(float* __restrict__ p, int n) {
  int i = blockIdx.x * blockDim.x + threadIdx.x;
  if (i < n) p[i] = 0.0f;
}

// ---------------------------------------------------------------------------
// Kernel 2: KV[bh] = k_m^T @ v  (per (b,h): [64 x T] @ [T x 64] -> [64 x 64])
// Split-K over T: 16 chunks of 256, accumulated with global f32 atomics.
// One wave per (bh, 16x16 tile, kchunk). Loads are coalesced (lanes span
// contiguous feature columns of one t-row of QKV); QKV rows stream from L2.
// ---------------------------------------------------------------------------
__global__ __launch_bounds__(256) void kv_accum(const float* __restrict__ QKV,
                                                float* __restrict__ KV) {
  const int lane = threadIdx.x & 31;
  const int wave = threadIdx.x >> 5;
  const int w = blockIdx.x * 8 + wave;       // 0..16383
  const int kchunk = w & 15;
  const int tile   = (w >> 4) & 15;
  const int bh     = w >> 8;                 // 0..63
  const int b = bh >> 4, h = bh & 15;

  const int d0 = (tile >> 2) * 16;           // output row tile (feature d)
  const int n0 = (tile & 3) * 16;            // output col tile (v feature m)
  const int half = lane >> 4;
  const int l    = lane & 15;

  const size_t rowbase = (size_t)b * SEQ;
  const int kcol = DMODEL     + h * HDIM;    // k_m slice columns
  const int vcol = 2 * DMODEL + h * HDIM;    // v slice columns

  v8f acc = {};
  const int t0 = kchunk * (SEQ / 16);
  for (int tt = 0; tt < SEQ / 16; tt += 4) {
    const int t = t0 + tt + 2 * half;
    const float* r0 = QKV + (rowbase + t)     * (size_t)QKV_N;
    const float* r1 = QKV + (rowbase + t + 1) * (size_t)QKV_N;
    v2f a, bb;
    a.x  = r0[kcol + d0 + l];                // A[d0+l][t]   = k_m[t][d0+l]
    a.y  = r1[kcol + d0 + l];                // A[d0+l][t+1]
    bb.x = r0[vcol + n0 + l];                // B[t][n0+l]   = v[t][n0+l]
    bb.y = r1[vcol + n0 + l];                // B[t+1][n0+l]
    acc = wmma_f32(a, bb, acc);
  }

  float* out = KV + (size_t)bh * HDIM * HDIM;
#pragma unroll
  for (int r = 0; r < 8; ++r) {
    const int d = d0 + r + 8 * half;
    atomicAdd(&out[d * HDIM + n0 + l], acc[r]);
  }
}

// ---------------------------------------------------------------------------
// Kernel 3: out = (q_m @ KV[bh]) / (sum_d q_m + 1e-6), written to ATT[B*T, C].
// One wave: 16 rows x full N=64 (4 WMMA accumulators sharing the A fragment).
// z fused: the A fragments streamed for WMMA *are* the q_m row, so per-lane
// partial sums + shfl_xor(16) + shfl broadcast recover z for free.
// grid = (T/128 = 32, B*H = 64), 8 waves / block.
// ---------------------------------------------------------------------------
__global__ __launch_bounds__(256) void attn_out(const float* __restrict__ QKV,
                                                const float* __restrict__ KV,
                                                float* __restrict__ ATT) {
  const int lane = threadIdx.x & 31;
  const int wave = threadIdx.x >> 5;
  const int half = lane >> 4;
  const int l    = lane & 15;

  const int bh = blockIdx.y;
  const int b = bh >> 4, h = bh & 15;
  const int t0 = blockIdx.x * 128 + wave * 16;
  const size_t rowbase = (size_t)b * SEQ + t0;

  const float* Arow = QKV + (rowbase + l) * (size_t)QKV_N + h * HDIM; // q_m row
  const float* kvp  = KV + (size_t)bh * HDIM * HDIM;

  v8f acc0 = {}, acc1 = {}, acc2 = {}, acc3 = {};
  float zpart = 0.0f;

#pragma unroll
  for (int k0 = 0; k0 < HDIM; k0 += 4) {
    const int ka = k0 + 2 * half;
    v2f a = *(const v2f*)(Arow + ka);
    zpart += a.x + a.y;                      // partial z for row t0+l
    const float* kr0 = kvp + ka * HDIM;
    const float* kr1 = kvp + (ka + 1) * HDIM;
    v2f b0, b1, b2, b3;
    b0.x = kr0[l];      b0.y = kr1[l];
    b1.x = kr0[16 + l]; b1.y = kr1[16 + l];
    b2.x = kr0[32 + l]; b2.y = kr1[32 + l];
    b3.x = kr0[48 + l]; b3.y = kr1[48 + l];
    acc0 = wmma_f32(a, b0, acc0);
    acc1 = wmma_f32(a, b1, acc1);
    acc2 = wmma_f32(a, b2, acc2);
    acc3 = wmma_f32(a, b3, acc3);
  }

  // Lanes l and l+16 hold complementary K-halves of z for row t0+l.
  zpart += __shfl_xor(zpart, 16, 32);

#pragma unroll
  for (int r = 0; r < 8; ++r) {
    const int rr = r + 8 * half;                       // C/D row within tile
    const float z = __shfl(zpart, rr, 32) + 1e-6f;     // lane rr holds z[row rr]
    const float inv = 1.0f / z;
    float* orow = ATT + (rowbase + rr) * (size_t)DMODEL + h * HDIM;
    orow[l]      = acc0[r] * inv;
    orow[16 + l] = acc1[r] * inv;
    orow[32 + l] = acc2[r] * inv;
    orow[48 + l] = acc3[r] * inv;
  }
}

// ---------------------------------------------------------------------------
extern "C" void kernel_launch(void* const* d_in, const int* in_sizes, int n_in,
                              void* d_out, int out_size, void* d_ws, size_t ws_size,
                              hipStream_t stream) {
  const float* x    = (const float*)d_in[0];
  const float* Wqkv = (const float*)d_in[1];
  const float* bqkv = (const float*)d_in[2];
  const float* Wout = (const float*)d_in[3];
  const float* bout = (const float*)d_in[4];
  float* out = (float*)d_out;

  // Workspace partition
  const size_t qkv_elems = (size_t)MROWS * QKV_N;                // 192 MB
  const size_t att_elems = (size_t)MROWS * DMODEL;               //  64 MB
  const size_t kv_elems  = (size_t)BATCH * NHEADS * HDIM * HDIM; //   1 MB
  float* QKV = (float*)d_ws;
  float* ATT = QKV + qkv_elems;
  float* KV  = ATT + att_elems;

  // 1) QKV projection + fused bias/feature-map epilogue
  gemm_staged<QKV_N, true><<<dim3(MROWS / 128, QKV_N / 64), 256, 0, stream>>>(
      x, Wqkv, bqkv, QKV);

  // 2) KV = k_m^T @ v  (split-K over T with f32 atomics)
  zero_f32<<<((int)kv_elems + 255) / 256, 256, 0, stream>>>(KV, (int)kv_elems);
  kv_accum<<<2048, 256, 0, stream>>>(QKV, KV);

  // 3) out = q_m @ KV, z-normalized
  attn_out<<<dim3(SEQ / 128, BATCH * NHEADS), 256, 0, stream>>>(QKV, KV, ATT);

  // 4) output projection
  gemm_staged<DMODEL, false><<<dim3(MROWS / 128, DMODEL / 64), 256, 0, stream>>>(
      ATT, Wout, bout, out);
}